// GCN_63161789055511
// MI455X (gfx1250) — compile-verified
//
#include <hip/hip_runtime.h>

#define D 128
#define EPS 1e-5f

typedef __attribute__((ext_vector_type(2))) float v2f;
typedef __attribute__((ext_vector_type(8))) float v8f;

// ---------------- utility kernels ----------------

__global__ void k_zero4(float4* __restrict__ p, long n4) {
    long i = (long)blockIdx.x * blockDim.x + threadIdx.x;
    if (i < n4) p[i] = make_float4(0.f, 0.f, 0.f, 0.f);
}

__global__ void k_count_deg(const int* __restrict__ dst, float* __restrict__ deg, int E) {
    int e = blockIdx.x * blockDim.x + threadIdx.x;
    if (e < E) atomicAdd(&deg[dst[e]], 1.0f);
}

__global__ void k_make_dinv(float* __restrict__ deg, int N) {
    int i = blockIdx.x * blockDim.x + threadIdx.x;
    if (i < N) deg[i] = __frsqrt_rn(deg[i] + 1.0f);   // +1 = self loop
}

// ---------------- fp32 WMMA GEMM: M = H(N x 128) @ W(128 x 128) ----------------
// block = 256 threads = 8 waves; blockIdx.x = 16-row tile; wave w -> 16-col tile w.
__global__ void k_gemm_wmma(const float* __restrict__ H, const float* __restrict__ W,
                            float* __restrict__ M, int nrows) {
    const int wave = threadIdx.x >> 5;
    const int lane = threadIdx.x & 31;
    const int half = lane >> 4;          // 0 | 1
    const int l16  = lane & 15;
    const int row0 = blockIdx.x * 16;
    const int col0 = wave * 16;

    int r = row0 + l16;
    if (r >= nrows) r = nrows - 1;       // clamp loads; EXEC must stay all-ones for WMMA
    const float* hrow = H + (size_t)r * D;

    v8f acc = {};
    #pragma unroll
    for (int k = 0; k < D; k += 4) {
        const int ka = k + half * 2;     // A: lanes 16-31 hold K+2 (ISA 16x4 layout)
        v2f a, b;
        a.x = hrow[ka];
        a.y = hrow[ka + 1];
        b.x = W[(size_t)ka * D + col0 + l16];        // B: 4x16, mirrored layout
        b.y = W[(size_t)(ka + 1) * D + col0 + l16];
        acc = __builtin_amdgcn_wmma_f32_16x16x4_f32(
                  false, a, false, b, (short)0, acc, false, false);
    }
    // C/D layout: VGPR v -> row (v + half*8), col l16
    #pragma unroll
    for (int v = 0; v < 8; ++v) {
        int mr = row0 + v + half * 8;
        if (mr < nrows) M[(size_t)mr * D + col0 + l16] = acc[v];
    }
}

// ---------------- edge scatter: agg[dst] += m[src] * dinv[src]*dinv[dst] -------
// one wave per edge, lane handles 4 consecutive feature dims (32*4 = 128).
__global__ void k_scatter(const int* __restrict__ src, const int* __restrict__ dst,
                          const float* __restrict__ Msg, const float* __restrict__ dinv,
                          float* __restrict__ agg, int E) {
    int e = (int)(((long)blockIdx.x * blockDim.x + threadIdx.x) >> 5);
    if (e >= E) return;
    const int lane = threadIdx.x & 31;
    const int s = src[e], d = dst[e];
    const float c = dinv[s] * dinv[d];
    float4 v = ((const float4*)(Msg + (size_t)s * D))[lane];
    float* a = agg + (size_t)d * D + lane * 4;
    atomicAdd(a + 0, v.x * c);
    atomicAdd(a + 1, v.y * c);
    atomicAdd(a + 2, v.z * c);
    atomicAdd(a + 3, v.w * c);
}

// ---------------- finalize: self-loop + bias (+ LayerNorm + ReLU) --------------
// one wave per node; 4 values per lane; wave32 shuffle reductions.
__global__ void k_finalize(const float* __restrict__ agg, const float* __restrict__ Msg,
                           const float* __restrict__ dinv, const float* __restrict__ bias,
                           const float* __restrict__ g, const float* __restrict__ be,
                           float* __restrict__ out, int N, int do_ln_relu) {
    int node = (int)(((long)blockIdx.x * blockDim.x + threadIdx.x) >> 5);
    if (node >= N) return;
    const int lane = threadIdx.x & 31;
    const size_t base = (size_t)node * D;

    const float di = dinv[node];
    const float c  = di * di;

    float4 a = ((const float4*)(agg + base))[lane];
    float4 m = ((const float4*)(Msg + base))[lane];
    float4 b = ((const float4*)bias)[lane];

    float v0 = a.x + m.x * c + b.x;
    float v1 = a.y + m.y * c + b.y;
    float v2 = a.z + m.z * c + b.z;
    float v3 = a.w + m.w * c + b.w;

    if (do_ln_relu) {
        float s = v0 + v1 + v2 + v3;
        #pragma unroll
        for (int off = 16; off > 0; off >>= 1) s += __shfl_xor(s, off, 32);
        const float mu = s * (1.0f / D);
        float d0 = v0 - mu, d1 = v1 - mu, d2 = v2 - mu, d3 = v3 - mu;
        float ss = d0 * d0 + d1 * d1 + d2 * d2 + d3 * d3;
        #pragma unroll
        for (int off = 16; off > 0; off >>= 1) ss += __shfl_xor(ss, off, 32);
        const float r = __frsqrt_rn(ss * (1.0f / D) + EPS);
        float4 gg = ((const float4*)g)[lane];
        float4 bb = ((const float4*)be)[lane];
        v0 = fmaxf(d0 * r * gg.x + bb.x, 0.f);
        v1 = fmaxf(d1 * r * gg.y + bb.y, 0.f);
        v2 = fmaxf(d2 * r * gg.z + bb.z, 0.f);
        v3 = fmaxf(d3 * r * gg.w + bb.w, 0.f);
    }
    ((float4*)(out + base))[lane] = make_float4(v0, v1, v2, v3);
}

// ---------------- host side ----------------

extern "C" void kernel_launch(void* const* d_in, const int* in_sizes, int n_in,
                              void* d_out, int out_size, void* d_ws, size_t ws_size,
                              hipStream_t stream) {
    const float* x  = (const float*)d_in[0];
    const int* edge = (const int*)d_in[1];
    const float* W1 = (const float*)d_in[2];  const float* b1 = (const float*)d_in[3];
    const float* W2 = (const float*)d_in[4];  const float* b2 = (const float*)d_in[5];
    const float* W3 = (const float*)d_in[6];  const float* b3 = (const float*)d_in[7];
    const float* g1 = (const float*)d_in[8];  const float* be1 = (const float*)d_in[9];
    const float* g2 = (const float*)d_in[10]; const float* be2 = (const float*)d_in[11];

    const int N = in_sizes[0] / D;
    const int E = in_sizes[1] / 2;
    const int* src = edge;
    const int* dst = edge + E;

    const int Npad = (N + 3) & ~3;
    float* dinv = (float*)d_ws;                     // N floats
    float* m    = dinv + Npad;                      // N*D floats (messages)
    float* agg  = m + (size_t)N * D;                // N*D floats (aggregate / h1)
    float* h2   = (float*)d_out;                    // output 0
    float* h3   = h2 + (size_t)N * D;               // output 1

    const long nd4 = (long)N * D / 4;
    const dim3 blk(256);
    const dim3 gZero((unsigned)((nd4 + 255) / 256));
    const dim3 gE((unsigned)((E + 255) / 256));
    const dim3 gN((unsigned)((N + 255) / 256));
    const dim3 gGemm((unsigned)((N + 15) / 16));
    const dim3 gEdgeW((unsigned)(((long)E * 32 + 255) / 256));
    const dim3 gNodeW((unsigned)(((long)N * 32 + 255) / 256));
    const dim3 gZeroN((unsigned)((Npad / 4 + 255) / 256));

    // degrees -> dinv
    k_zero4<<<gZeroN, blk, 0, stream>>>((float4*)dinv, Npad / 4);
    k_count_deg<<<gE, blk, 0, stream>>>(dst, dinv, E);
    k_make_dinv<<<gN, blk, 0, stream>>>(dinv, N);

    // ---- layer 1: h1 = relu(LN(conv(x, W1, b1)))  (h1 stored in-place in agg) ----
    k_gemm_wmma<<<gGemm, blk, 0, stream>>>(x, W1, m, N);
    k_zero4<<<gZero, blk, 0, stream>>>((float4*)agg, nd4);
    k_scatter<<<gEdgeW, blk, 0, stream>>>(src, dst, m, dinv, agg, E);
    k_finalize<<<gNodeW, blk, 0, stream>>>(agg, m, dinv, b1, g1, be1, agg, N, 1);

    // ---- layer 2: h2 = relu(LN(conv(h1, W2, b2))) -> d_out[0 : N*D] ----
    k_gemm_wmma<<<gGemm, blk, 0, stream>>>(agg, W2, m, N);   // reads h1 before zero
    k_zero4<<<gZero, blk, 0, stream>>>((float4*)agg, nd4);
    k_scatter<<<gEdgeW, blk, 0, stream>>>(src, dst, m, dinv, agg, E);
    k_finalize<<<gNodeW, blk, 0, stream>>>(agg, m, dinv, b2, g2, be2, h2, N, 1);

    // ---- layer 3: h3 = conv(h2, W3, b3) -> d_out[N*D : 2*N*D] ----
    k_gemm_wmma<<<gGemm, blk, 0, stream>>>(h2, W3, m, N);
    k_zero4<<<gZero, blk, 0, stream>>>((float4*)agg, nd4);
    k_scatter<<<gEdgeW, blk, 0, stream>>>(src, dst, m, dinv, agg, E);
    k_finalize<<<gNodeW, blk, 0, stream>>>(agg, m, dinv, b3, b3, b3, h3, N, 0);
}